// VectorQuantizer_84095459656194
// MI455X (gfx1250) — compile-verified
//
#include <hip/hip_runtime.h>
#include <hip/hip_bf16.h>
#include <math.h>

// Problem constants (from reference):
//   z: (4, 256, 16, 32, 32) f32 ; embedding: (1024, 256) f32
//   N = 4*16*32*32 = 65536 rows, D = 256 channels, K = 1024 codes
#define KCODES 1024
#define DCH    256
#define NROWS  65536
#define PER_B  16384          // Dd*H*W : per-batch row count, also channel stride in z
#define BETA_C 0.25f

typedef __attribute__((ext_vector_type(2))) float v2f;
typedef __attribute__((ext_vector_type(8))) float v8f;

// ---------------------------------------------------------------------------
// Kernel 1: per-code squared norms + zero the histogram (d_ws is poisoned).
// One wave (32 lanes) per code.
// ---------------------------------------------------------------------------
__global__ __launch_bounds__(32) void vq_norms(const float* __restrict__ emb,
                                               float* __restrict__ normE,
                                               unsigned* __restrict__ hist) {
  const int code = blockIdx.x;
  const int lane = threadIdx.x;
  const float* row = emb + (size_t)code * DCH;
  float s = 0.f;
#pragma unroll
  for (int j = 0; j < DCH; j += 32) {
    float v = row[j + lane];
    s += v * v;
  }
#pragma unroll
  for (int m = 16; m >= 1; m >>= 1) s += __shfl_xor(s, m, 32);
  if (lane == 0) {
    normE[code] = s;
    hist[code]  = 0u;
  }
}

// ---------------------------------------------------------------------------
// Kernel 2: distance GEMM + argmin via V_WMMA_F32_16X16X4_F32.
// Block = 256 threads = 8 waves; each wave owns one 16-row tile of flat(z).
// A (16x4 f32) fragments for the whole D=256 are preloaded into registers
// (64 x v2f). Loop over 64 code-tiles of 16 codes staged in LDS.
// dist = ||e||^2 - 2 * (x . e)   (row-constant ||x||^2 dropped — argmin-safe)
// ---------------------------------------------------------------------------
__global__ __launch_bounds__(256) void vq_argmin(const float* __restrict__ z,
                                                 const float* __restrict__ emb,
                                                 const float* __restrict__ normE,
                                                 unsigned* __restrict__ hist,
                                                 int* __restrict__ idxbuf,
                                                 float* __restrict__ idx_out_f) {
  __shared__ float sh[16 * 260];        // 16 codes x 256 ch, padded stride 260

  const int tid  = threadIdx.x;
  const int wave = tid >> 5;
  const int lane = tid & 31;
  const int half = lane >> 4;           // 0: lanes 0-15, 1: lanes 16-31
  const int m    = lane & 15;

  const int tile = blockIdx.x * 8 + wave;   // 0..4095
  const int n0   = tile << 4;               // first row of this 16-row tile
  const int b    = n0 >> 14;                // batch (tile never crosses batch)
  const int rem0 = n0 & 16383;              // flat (dd,h,w) offset within batch

  // A-fragment layout (16x4 f32): lanes 0-15 hold M=0..15 with K=0 (v.x),K=1 (v.y);
  // lanes 16-31 hold K=2 (v.x), K=3 (v.y).  flat[n][c] = z[(b*256+c)*16384 + rem].
  const float* zb = z + ((size_t)b << 22) + (size_t)(rem0 + m);
  v2f areg[64];
#pragma unroll
  for (int s = 0; s < 64; ++s) {
    const int c = 4 * s + 2 * half;
    areg[s].x = zb[(size_t)c * PER_B];
    areg[s].y = zb[(size_t)(c + 1) * PER_B];
  }

  float minv[8];
  int   mini[8];
#pragma unroll
  for (int g = 0; g < 8; ++g) { minv[g] = 3.4e38f; mini[g] = 0; }

  for (int kt = 0; kt < 64; ++kt) {
    __syncthreads();
    // Cooperative stage of 16 codes x 256 channels (coalesced, uniform loop).
    const float* esrc = emb + (size_t)(kt << 4) * DCH;
    for (int i = tid; i < 16 * DCH; i += 256) {
      const int code = i >> 8, ch = i & 255;
      sh[code * 260 + ch] = esrc[i];
    }
    __syncthreads();

    // B-fragment (4x16 f32): lanes hold N=code=lane%16; v.x/v.y = K pair per half.
    v8f acc = {};
#pragma unroll
    for (int s = 0; s < 64; ++s) {
      v2f bf = *(const v2f*)&sh[m * 260 + 4 * s + 2 * half];
      acc = __builtin_amdgcn_wmma_f32_16x16x4_f32(false, areg[s], false, bf,
                                                  (short)0, acc, false, false);
    }

    // C/D layout: VGPR g holds row (g + 8*half), column N = lane%16.
    const int   code = (kt << 4) + m;
    const float nrm  = normE[code];
#pragma unroll
    for (int g = 0; g < 8; ++g) {
      const float d = nrm - 2.0f * acc[g];
      const bool better = d < minv[g];          // strict < : lowest code wins ties
      minv[g] = better ? d : minv[g];
      mini[g] = better ? code : mini[g];
    }
  }

  // Cross-lane argmin within each 16-lane half (rows stay separated, wave32).
#pragma unroll
  for (int g = 0; g < 8; ++g) {
    float v = minv[g];
    int   ii = mini[g];
#pragma unroll
    for (int mm = 1; mm <= 8; mm <<= 1) {
      const float ov = __shfl_xor(v, mm, 32);
      const int   oi = __shfl_xor(ii, mm, 32);
      const bool take = (ov < v) || (ov == v && oi < ii);
      v  = take ? ov : v;
      ii = take ? oi : ii;
    }
    minv[g] = v;
    mini[g] = ii;
  }

  if (m == 0) {                 // lane 0 writes rows 0-7, lane 16 writes rows 8-15
#pragma unroll
    for (int g = 0; g < 8; ++g) {
      const int n   = n0 + (half << 3) + g;
      const int idx = mini[g];
      idxbuf[n]    = idx;
      idx_out_f[n] = (float)idx;
      atomicAdd(&hist[idx], 1u);
    }
  }
}

// ---------------------------------------------------------------------------
// Kernel 3: z_q gather in output (B,C,Dd,H,W) layout + per-block loss partials.
// i = (b*256+c)*16384 + rem ; n = b*16384 + rem.  Coalesced over rem.
// ---------------------------------------------------------------------------
__global__ __launch_bounds__(256) void vq_gather(const float* __restrict__ z,
                                                 const float* __restrict__ emb,
                                                 const int* __restrict__ idxbuf,
                                                 float* __restrict__ zq_out,
                                                 float* __restrict__ blocksums) {
  __shared__ float red[256];
  const size_t i = (size_t)blockIdx.x * 256 + threadIdx.x;   // < 2^24
  const int c   = (int)((i >> 14) & 255);
  const int b   = (int)(i >> 22);
  const int rem = (int)(i & 16383);
  const int n   = (b << 14) | rem;

  const float e = emb[(size_t)idxbuf[n] * DCH + c];
  zq_out[i] = e;                      // straight-through output == z_q numerically
  const float d = z[i] - e;
  red[threadIdx.x] = d * d;
  __syncthreads();
  for (int s = 128; s > 0; s >>= 1) {
    if (threadIdx.x < (unsigned)s) red[threadIdx.x] += red[threadIdx.x + s];
    __syncthreads();
  }
  if (threadIdx.x == 0) blocksums[blockIdx.x] = red[0];
}

// ---------------------------------------------------------------------------
// Kernel 4: deterministic final reductions -> vq_loss and perplexity.
// ---------------------------------------------------------------------------
__global__ __launch_bounds__(256) void vq_finalize(const float* __restrict__ blocksums,
                                                   const unsigned* __restrict__ hist,
                                                   float* __restrict__ loss_out,
                                                   float* __restrict__ perp_out) {
  __shared__ float red[256];
  const int t = threadIdx.x;

  float s = 0.f;
  for (int j = t; j < 65536; j += 256) s += blocksums[j];
  red[t] = s;
  __syncthreads();
  for (int st = 128; st > 0; st >>= 1) {
    if (t < st) red[t] += red[t + st];
    __syncthreads();
  }
  if (t == 0)
    loss_out[0] = BETA_C * red[0] / (float)((size_t)NROWS * DCH);
  __syncthreads();

  float h = 0.f;
  for (int j = t; j < KCODES; j += 256) {
    const float p = (float)hist[j] / (float)NROWS;
    h += p * logf(p + 1e-10f);
  }
  red[t] = h;
  __syncthreads();
  for (int st = 128; st > 0; st >>= 1) {
    if (t < st) red[t] += red[t + st];
    __syncthreads();
  }
  if (t == 0) perp_out[0] = expf(-red[0]);
}

// ---------------------------------------------------------------------------
extern "C" void kernel_launch(void* const* d_in, const int* in_sizes, int n_in,
                              void* d_out, int out_size, void* d_ws, size_t ws_size,
                              hipStream_t stream) {
  const float* z   = (const float*)d_in[0];   // (4,256,16,32,32)
  const float* emb = (const float*)d_in[1];   // (1024,256)
  float* out = (float*)d_out;

  // Workspace layout (floats): normE[1024] | hist[1024] | idx[65536] | bsums[65536]
  float*    ws    = (float*)d_ws;
  float*    normE = ws;
  unsigned* hist  = (unsigned*)(ws + 1024);
  int*      idxb  = (int*)(ws + 2048);
  float*    bsums = ws + 2048 + 65536;

  // Output layout (floats): z_q[16777216] | vq_loss | indices[65536] | perplexity
  float* zq   = out;
  float* loss = out + 16777216;
  float* idxf = out + 16777217;
  float* perp = out + 16777217 + 65536;

  vq_norms  <<<KCODES, 32, 0, stream>>>(emb, normE, hist);
  vq_argmin <<<512, 256, 0, stream>>>(z, emb, normE, hist, idxb, idxf);
  vq_gather <<<65536, 256, 0, stream>>>(z, emb, idxb, zq, bsums);
  vq_finalize<<<1, 256, 0, stream>>>(bsums, hist, loss, perp);
}